// SAGE_86285892977010
// MI455X (gfx1250) — compile-verified
//
#include <hip/hip_runtime.h>
#include <hip/hip_bf16.h>

#define N_NODES 50000
#define EH_EDGES 500000
#define ES_EDGES 250000
#define D 128        // D_IN == D_HID
#define NCLS 47

typedef __attribute__((ext_vector_type(2))) float v2f;
typedef __attribute__((ext_vector_type(8))) float v8f;

// ---------------------------------------------------------------- zero fill
__global__ void zero_f4(float4* __restrict__ p, long n4) {
  long i = (long)blockIdx.x * blockDim.x + threadIdx.x;
  long stride = (long)gridDim.x * blockDim.x;
  float4 z = {0.f, 0.f, 0.f, 0.f};
  for (; i < n4; i += stride) p[i] = z;
}

// -------------------------------------------- scatter-sum of feat[src] -> dst
// one wave32 per edge; each lane moves float4 (32*16B = full 512B row)
__global__ void scatter_hist(const float* __restrict__ feat,
                             const int* __restrict__ src,
                             const int* __restrict__ dst,
                             float* __restrict__ sum,
                             float* __restrict__ cnt, int E) {
  int e = (int)((blockIdx.x * blockDim.x + threadIdx.x) >> 5);
  int lane = threadIdx.x & 31;
  if (e >= E) return;
  int s = src[e], d = dst[e];
  float4 v = ((const float4*)(feat + (size_t)s * D))[lane];
  float* o = sum + (size_t)d * D + lane * 4;
  unsafeAtomicAdd(o + 0, v.x);
  unsafeAtomicAdd(o + 1, v.y);
  unsafeAtomicAdd(o + 2, v.z);
  unsafeAtomicAdd(o + 3, v.w);
  if (lane == 0) unsafeAtomicAdd(cnt + d, 1.0f);
}

// ----------------------------- scatter-sum of (h - hbar)[src] -> dst (delta)
__global__ void scatter_delta(const float* __restrict__ h,
                              const float* __restrict__ hbar,
                              const int* __restrict__ src,
                              const int* __restrict__ dst,
                              float* __restrict__ sum,
                              float* __restrict__ cnt, int E) {
  int e = (int)((blockIdx.x * blockDim.x + threadIdx.x) >> 5);
  int lane = threadIdx.x & 31;
  if (e >= E) return;
  int s = src[e], d = dst[e];
  float4 a = ((const float4*)(h    + (size_t)s * D))[lane];
  float4 b = ((const float4*)(hbar + (size_t)s * D))[lane];
  float* o = sum + (size_t)d * D + lane * 4;
  unsafeAtomicAdd(o + 0, a.x - b.x);
  unsafeAtomicAdd(o + 1, a.y - b.y);
  unsafeAtomicAdd(o + 2, a.z - b.z);
  unsafeAtomicAdd(o + 3, a.w - b.w);
  if (lane == 0) unsafeAtomicAdd(cnt + d, 1.0f);
}

// -------- h_neigh = sum_h/max(cnt_h,1) + sum_s/max(cnt_s,1), in-place sum_h
__global__ void finalize_hneigh(float* __restrict__ sum_h,
                                const float* __restrict__ sum_s,
                                const float* __restrict__ cnt_h,
                                const float* __restrict__ cnt_s, long n) {
  long i = (long)blockIdx.x * blockDim.x + threadIdx.x;
  if (i >= n) return;
  int node = (int)(i >> 7);  // / 128
  float ch = fmaxf(cnt_h[node], 1.0f);
  float cs = fmaxf(cnt_s[node], 1.0f);
  sum_h[i] = sum_h[i] / ch + sum_s[i] / cs;
}

// ---------------------------------------------------------------- WMMA GEMM
// out[m, j] = relu( sum_k H[m,k]*W[j,k] + sum_k G[m,k]*W[j,128+k] + bias[j] )
// Block = 256 threads = 8 waves, one 16-row M-tile per block, one 16-col
// N-tile per wave. A tile (16x256 f32) staged in LDS, row pad +4 floats so
// the A-operand pattern At[lane&15][kg + 2*(lane>>4)] is bank-conflict-free.
template <int OUT, int NT>
__global__ __launch_bounds__(256) void sage_linear_relu(
    const float* __restrict__ H, const float* __restrict__ G,
    const float* __restrict__ W, const float* __restrict__ bias,
    float* __restrict__ out) {
  __shared__ float At[16][260];  // 16 rows x 256 cols, +4 pad
  const int mbase = blockIdx.x * 16;
  const int t = threadIdx.x;

  // cooperative load of A tile: 16 threads per row, 16 cols (4 float4) each
  {
    int row = t >> 4;
    int cbase = (t & 15) * 16;  // 0..240, chunk fully inside H or G half
    const float* srow = (cbase < D)
                            ? (H + (size_t)(mbase + row) * D + cbase)
                            : (G + (size_t)(mbase + row) * D + (cbase - D));
#pragma unroll
    for (int i = 0; i < 4; ++i) {
      float4 v = *(const float4*)(srow + i * 4);
      *(float4*)&At[row][cbase + i * 4] = v;
    }
  }
  __syncthreads();

  const int wv = t >> 5;
  if (wv >= NT) return;  // wave-uniform: active waves keep EXEC all-ones
  const int lane = t & 31;
  const int nn = lane & 15;          // A row (M) and B col (N) for this lane
  const int koff = (lane >> 4) * 2;  // lanes 16-31 hold K=2,3 of each step
  const int j = wv * 16 + nn;        // output channel
  const bool jok = (j < OUT);
  const float* wrow = W + (size_t)j * (2 * D);

  v8f c = {0.f, 0.f, 0.f, 0.f, 0.f, 0.f, 0.f, 0.f};
#pragma unroll 8
  for (int kg = 0; kg < 2 * D; kg += 4) {
    float2 av = *(const float2*)&At[nn][kg + koff];  // ds_load_b64
    v2f a; a.x = av.x; a.y = av.y;
    v2f b; b.x = 0.f; b.y = 0.f;
    if (jok) {
      float2 bv = *(const float2*)(wrow + kg + koff);
      b.x = bv.x; b.y = bv.y;
    }
    // D = A(16x4) x B(4x16) + C, full f32
    c = __builtin_amdgcn_wmma_f32_16x16x4_f32(
        /*neg_a=*/false, a, /*neg_b=*/false, b,
        /*c_mod=*/(short)0, c, /*reuse_a=*/false, /*reuse_b=*/false);
  }

  if (jok) {
    float bj = bias[j];
    int mrow = mbase + ((lane >> 4) << 3);  // lanes 16-31 hold M = r+8
#pragma unroll
    for (int r = 0; r < 8; ++r) {
      float v = c[r] + bj;
      out[(size_t)(mrow + r) * OUT + j] = fmaxf(v, 0.f);
    }
  }
}

// ---------------------------------------------------------------------------
extern "C" void kernel_launch(void* const* d_in, const int* in_sizes, int n_in,
                              void* d_out, int out_size, void* d_ws, size_t ws_size,
                              hipStream_t stream) {
  const float* x     = (const float*)d_in[0];
  const float* hbar0 = (const float*)d_in[1];
  const float* hbar1 = (const float*)d_in[2];
  const float* W0    = (const float*)d_in[3];
  const float* b0    = (const float*)d_in[4];
  const float* W1    = (const float*)d_in[5];
  const float* b1    = (const float*)d_in[6];
  const int* hs0 = (const int*)d_in[7];
  const int* hd0 = (const int*)d_in[8];
  const int* ss0 = (const int*)d_in[9];
  const int* sd0 = (const int*)d_in[10];
  const int* hs1 = (const int*)d_in[11];
  const int* hd1 = (const int*)d_in[12];
  const int* ss1 = (const int*)d_in[13];
  const int* sd1 = (const int*)d_in[14];

  // workspace layout (floats): h0 | sum_h | sum_s | cnt_h | cnt_s
  float* h0    = (float*)d_ws;
  float* sum_h = h0 + (size_t)N_NODES * D;
  float* sum_s = sum_h + (size_t)N_NODES * D;
  float* cnt_h = sum_s + (size_t)N_NODES * D;
  float* cnt_s = cnt_h + N_NODES;

  const long accN  = 2L * N_NODES * D + 2L * N_NODES;  // sum_h..cnt_s span
  const long accN4 = accN / 4;                         // divisible by 4
  const long nel   = (long)N_NODES * D;
  const int hist_blocks = (EH_EDGES * 32 + 255) / 256;
  const int samp_blocks = (ES_EDGES * 32 + 255) / 256;
  const int elem_blocks = (int)((nel + 255) / 256);
  const int m_tiles = N_NODES / 16;  // 3125, exact

  // ---- layer 0
  zero_f4<<<2048, 256, 0, stream>>>((float4*)sum_h, accN4);
  scatter_hist<<<hist_blocks, 256, 0, stream>>>(hbar0, hs0, hd0, sum_h, cnt_h, EH_EDGES);
  scatter_delta<<<samp_blocks, 256, 0, stream>>>(x, hbar0, ss0, sd0, sum_s, cnt_s, ES_EDGES);
  finalize_hneigh<<<elem_blocks, 256, 0, stream>>>(sum_h, sum_s, cnt_h, cnt_s, nel);
  sage_linear_relu<D, 8><<<m_tiles, 256, 0, stream>>>(x, sum_h, W0, b0, h0);

  // ---- layer 1
  zero_f4<<<2048, 256, 0, stream>>>((float4*)sum_h, accN4);
  scatter_hist<<<hist_blocks, 256, 0, stream>>>(hbar1, hs1, hd1, sum_h, cnt_h, EH_EDGES);
  scatter_delta<<<samp_blocks, 256, 0, stream>>>(h0, hbar1, ss1, sd1, sum_s, cnt_s, ES_EDGES);
  finalize_hneigh<<<elem_blocks, 256, 0, stream>>>(sum_h, sum_s, cnt_h, cnt_s, nel);
  sage_linear_relu<NCLS, 3><<<m_tiles, 256, 0, stream>>>(h0, sum_h, W1, b1, (float*)d_out);
}